// SongsFeatureTraining_79534204387545
// MI455X (gfx1250) — compile-verified
//
#include <hip/hip_runtime.h>
#include <math.h>

// Tanh RNN, fused persistent kernel.  T=2048 B=256 I=16 H=64.
// Grid: 16 workgroups (one 16-row batch tile each) x 128 threads (4 wave32).
// Wave w owns hidden columns [16w, 16w+16).  h ping-pongs in LDS (1 barrier
// per step); weights live in per-wave f32 WMMA fragments.  Matrix math via
// V_WMMA_F32_16X16X4_F32; pointwise tanh via hardware V_TANH_F32.

typedef float v2f __attribute__((ext_vector_type(2)));
typedef float v8f __attribute__((ext_vector_type(8)));

#define WMMA_F32(A, Bf, C) \
  __builtin_amdgcn_wmma_f32_16x16x4_f32(false, (A), false, (Bf), (short)0, (C), false, false)

// Hardware tanh.  Dispatch happens per compilation pass: the device pass uses
// the gfx1250 builtin when declared (else raw V_TANH_F32 asm); the host pass
// parses the asm branch but never codegens this __device__ body.
__device__ __forceinline__ float fast_tanh(float x) {
#if __has_builtin(__builtin_amdgcn_tanhf)
  return __builtin_amdgcn_tanhf(x);
#else
  float r;
  asm volatile("v_tanh_f32 %0, %1" : "=v"(r) : "v"(x));
  return r;
#endif
}

__global__ __launch_bounds__(128) void rnn_tanh_wmma_f32(
    const float* __restrict__ x,      // [T, B, I]
    const float* __restrict__ w_ih,   // [H, I]
    const float* __restrict__ w_hh,   // [H, H]
    const float* __restrict__ b_ih,   // [H]
    const float* __restrict__ b_hh,   // [H]
    float* __restrict__ out)          // [T,B,H] then [1,B,H]
{
  constexpr int T = 2048, B = 256, I = 16, H = 64;
  constexpr int LPAD = 68;                      // padded LDS row (bank-conflict free)
  constexpr int HBUF = 16 * LPAD;

  __shared__ float h_lds[2 * HBUF];             // ping-pong h buffers

  const int lane  = threadIdx.x & 31;
  const int wave  = threadIdx.x >> 5;           // 0..3 -> hidden N-tile
  const int nbase = wave * 16;
  const int b0    = blockIdx.x * 16;            // batch tile base

  const int lrow  = lane & 15;                  // M (or N) index inside a 16-tile
  const int khalf = (lane >> 4) << 1;           // 0 for lanes 0-15, 2 for lanes 16-31
  const int rbase = (lane >> 4) << 3;           // C/D: rows 0-7 vs 8-15

  // ---- preload W_hh^T B-fragments: frag f covers K = 4f..4f+3 of the 64-deep dot ----
  v2f bhh[16];
#pragma unroll
  for (int f = 0; f < 16; ++f)
    bhh[f] = *(const v2f*)(w_hh + (nbase + lrow) * H + (4 * f + khalf));

  // ---- preload W_ih^T B-fragments (K = I = 16) ----
  v2f bih[4];
#pragma unroll
  for (int f = 0; f < 4; ++f)
    bih[f] = *(const v2f*)(w_ih + (nbase + lrow) * I + (4 * f + khalf));

  // ---- bias fragment: column n = nbase+lrow, replicated over the M dimension ----
  const float bias = b_ih[nbase + lrow] + b_hh[nbase + lrow];
  v8f cbias;
#pragma unroll
  for (int r = 0; r < 8; ++r) cbias[r] = bias;

  // ---- h0 = 0 (both buffers; t=0 reads buffer 0) ----
  for (int i = threadIdx.x; i < 2 * HBUF; i += 128) h_lds[i] = 0.0f;
  __syncthreads();

  const int orow = b0 + rbase;                  // output batch-row base for this lane
  const int ocol = nbase + lrow;                // output hidden column for this lane

  v8f acc;
  for (int t = 0; t < T; ++t) {
    const float* hr = h_lds + (t & 1) * HBUF;        // h_t
    float*       hw = h_lds + ((t + 1) & 1) * HBUF;  // h_{t+1}

    // x A-fragments for this step (K = I = 16); contiguous 8B per lane
    const float* xr = x + ((size_t)t * B + b0 + lrow) * I + khalf;
    v2f ax[4];
#pragma unroll
    for (int f = 0; f < 4; ++f) ax[f] = *(const v2f*)(xr + 4 * f);
    // prefetch next step's x tile (timestep clamped -> no branch)
    const int tn = (t + 1 < T) ? (t + 1) : t;
    __builtin_prefetch(x + ((size_t)tn * B + b0 + lrow) * I + khalf, 0, 1);

    // h A-fragments from LDS (K = H = 64)
    v2f ah[16];
#pragma unroll
    for (int f = 0; f < 16; ++f)
      ah[f] = *(const v2f*)(hr + lrow * LPAD + 4 * f + khalf);

    // D = bias + x@W_ih^T + h@W_hh^T; 4 round-robin accumulators to keep
    // same-accumulator WMMA dependencies 4 instructions apart.
    v8f a0 = cbias;
    v8f a1 = {0.f, 0.f, 0.f, 0.f, 0.f, 0.f, 0.f, 0.f};
    v8f a2 = a1, a3 = a1;
    a0 = WMMA_F32(ax[0], bih[0], a0);
    a1 = WMMA_F32(ax[1], bih[1], a1);
    a2 = WMMA_F32(ax[2], bih[2], a2);
    a3 = WMMA_F32(ax[3], bih[3], a3);
#pragma unroll
    for (int f = 0; f < 4; ++f) {
      a0 = WMMA_F32(ah[4 * f + 0], bhh[4 * f + 0], a0);
      a1 = WMMA_F32(ah[4 * f + 1], bhh[4 * f + 1], a1);
      a2 = WMMA_F32(ah[4 * f + 2], bhh[4 * f + 2], a2);
      a3 = WMMA_F32(ah[4 * f + 3], bhh[4 * f + 3], a3);
    }
#pragma unroll
    for (int r = 0; r < 8; ++r)
      acc[r] = fast_tanh((a0[r] + a1[r]) + (a2[r] + a3[r]));  // hw V_TANH_F32

    // publish h_{t+1} to the other LDS buffer and stream the tile to out[t]
#pragma unroll
    for (int r = 0; r < 8; ++r)
      hw[(rbase + r) * LPAD + ocol] = acc[r];
    float* op = out + ((size_t)t * B + orow) * H + ocol;
#pragma unroll
    for (int r = 0; r < 8; ++r)
      op[(size_t)r * H] = acc[r];

    // Single barrier per step: reads of hr happened above; the only writes to
    // hr occur in the *next* iteration, after this barrier.
    __syncthreads();
  }

  // h_last tail: [1, B, H] appended after the [T, B, H] output
  float* hl = out + (size_t)T * B * H + (size_t)orow * H + ocol;
#pragma unroll
  for (int r = 0; r < 8; ++r) hl[(size_t)r * H] = acc[r];
}

extern "C" void kernel_launch(void* const* d_in, const int* in_sizes, int n_in,
                              void* d_out, int out_size, void* d_ws, size_t ws_size,
                              hipStream_t stream) {
  (void)in_sizes; (void)n_in; (void)d_ws; (void)ws_size; (void)out_size;
  const float* x    = (const float*)d_in[0];
  const float* w_ih = (const float*)d_in[1];
  const float* w_hh = (const float*)d_in[2];
  const float* b_ih = (const float*)d_in[3];
  const float* b_hh = (const float*)d_in[4];
  float* out = (float*)d_out;

  dim3 grid(256 / 16);   // 16 batch tiles
  dim3 block(128);       // 4 wave32 per workgroup
  rnn_tanh_wmma_f32<<<grid, block, 0, stream>>>(x, w_ih, w_hh, b_ih, b_hh, out);
}